// PMALayer_62483184222839
// MI455X (gfx1250) — compile-verified
//
#include <hip/hip_runtime.h>

typedef __attribute__((ext_vector_type(2))) float v2f;
typedef __attribute__((ext_vector_type(8))) float v8f;

#define HIDN 512
#define NHD  8
#define DHD  64
#define NSEQ 512
#define NB   128
#define AST  520   // alpha LDS row stride (floats) -> banks (8*row + n) % 64, conflict-free

// ---------------------------------------------------------------------------
// Kernel A (tiny, 1 block): query = w_seed_w*seed + w_seed_b ; q = w_q @ query
// qt[h][j] = (1/8) * sum_d q[h*64+d] * w_v[(h*64+d)*512 + j]
// ---------------------------------------------------------------------------
__global__ void __launch_bounds__(256)
prep_qt_kernel(const float* __restrict__ w_q, const float* __restrict__ w_v,
               const float* __restrict__ w_seed_w, const float* __restrict__ w_seed_b,
               const float* __restrict__ seed_vec, float* __restrict__ qt) {
    __shared__ float query[HIDN];
    __shared__ float qv[HIDN];
    const int tid = threadIdx.x;
    const float sv = seed_vec[0];
    for (int i = tid; i < HIDN; i += 256) query[i] = w_seed_w[i] * sv + w_seed_b[i];
    __syncthreads();
    for (int i = tid; i < HIDN; i += 256) {
        float s = 0.f;
        const float* r = w_q + (size_t)i * HIDN;
        for (int j = 0; j < HIDN; ++j) s += r[j] * query[j];
        qv[i] = s;
    }
    __syncthreads();
    for (int idx = tid; idx < NHD * HIDN; idx += 256) {
        const int hd = idx >> 9;
        const int j  = idx & (HIDN - 1);
        float s = 0.f;
        for (int d = 0; d < DHD; ++d)
            s += qv[hd * DHD + d] * w_v[(size_t)(hd * DHD + d) * HIDN + j];
        qt[idx] = s * 0.125f;   // fold 1/sqrt(DH)
    }
}

// ---------------------------------------------------------------------------
// Kernel B (one block per batch element b, 256 threads = 8 waves):
//   Phase 1: e[n,h]   = h[b,n,:] . qt[h,:]        (WMMA, M=n-tiles, Ncols=16 with
//                                                  cols 8..15 duplicated & ignored)
//   Phase 2: masked softmax per head (wave h handles head h), write alpha
//   Phase 3: ctx[h,j] = sum_n alpha[h,n]*h[b,n,j]  (WMMA, rows 8..15 duplicated & ignored)
// ---------------------------------------------------------------------------
__global__ void __launch_bounds__(256)
pma_core_kernel(const float* __restrict__ hmat, const int* __restrict__ lengths,
                const float* __restrict__ qt, float* __restrict__ alpha_out,
                float* __restrict__ ctx) {
    __shared__ float qtT[HIDN * NHD];   // [k][head], 16 KB
    __shared__ float aL[NHD * AST];     // logits -> alpha per head, ~16.6 KB

    const int b    = blockIdx.x;
    const int tid  = threadIdx.x;
    const int lane = tid & 31;
    const int wv   = tid >> 5;
    const int len  = lengths[b];
    const int row  = lane & 15;
    const int kh   = lane >> 4;
    const float* hb = hmat + (size_t)b * NSEQ * HIDN;

    // stage qtT[k*8 + h] = qt[h*512 + k]
    for (int i = tid; i < HIDN * NHD; i += 256) {
        const int k = i >> 3, hd = i & 7;
        qtT[i] = qt[hd * HIDN + k];
    }
    __syncthreads();

    // ---- Phase 1: logits via WMMA (each wave: 4 n-tiles of 16) ----
    for (int t = 0; t < 4; ++t) {
        const int m0 = (wv * 4 + t) * 16;
        v8f acc = {};
        const float* ap = hb + (size_t)(m0 + row) * HIDN + 2 * kh;
        const float* bp = qtT + (lane & 7);   // cols 8..15 duplicate cols 0..7 (outputs unused)
        for (int k = 0; k < HIDN; k += 4) {
            v2f a = *(const v2f*)(ap + k);
            v2f bb;
            bb.x = bp[(k + 2 * kh) * 8];
            bb.y = bp[(k + 2 * kh + 1) * 8];
            acc = __builtin_amdgcn_wmma_f32_16x16x4_f32(false, a, false, bb,
                                                        (short)0, acc, false, false);
        }
        if ((lane & 15) < 8) {   // only real head columns
            const int hd = lane & 7;
            #pragma unroll
            for (int r = 0; r < 8; ++r) {
                const int n = m0 + r + 8 * kh;
                aL[hd * AST + n] = (n < len) ? acc[r] : -__builtin_inff();
            }
        }
    }
    __syncthreads();

    // ---- Phase 2: softmax, wave wv == head wv ----
    {
        const int hd = wv;
        float mx = -__builtin_inff();
        for (int n = lane; n < NSEQ; n += 32) mx = fmaxf(mx, aL[hd * AST + n]);
        for (int off = 16; off > 0; off >>= 1) mx = fmaxf(mx, __shfl_xor(mx, off, 32));
        float s = 0.f;
        for (int n = lane; n < NSEQ; n += 32) {
            const float ex = __expf(aL[hd * AST + n] - mx);  // -inf rows -> 0
            aL[hd * AST + n] = ex;
            s += ex;
        }
        for (int off = 16; off > 0; off >>= 1) s += __shfl_xor(s, off, 32);
        const float inv = 1.f / s;
        float* aout = alpha_out + ((size_t)b * NHD + hd) * NSEQ;
        for (int n = lane; n < NSEQ; n += 32) {
            const float a = aL[hd * AST + n] * inv;
            aL[hd * AST + n] = a;
            aout[n] = a;
        }
    }
    __syncthreads();

    // ---- Phase 3: ctx via WMMA (each wave: 4 j-tiles of 16) ----
    for (int t = 0; t < 4; ++t) {
        const int j0 = (wv * 4 + t) * 16;
        v8f acc = {};
        const float* ap = aL + (size_t)(row & 7) * AST + 2 * kh; // rows 8..15 dup, ignored
        const float* bp = hb + (size_t)(2 * kh) * HIDN + j0 + row;
        for (int n = 0; n < NSEQ; n += 4) {
            v2f a = *(const v2f*)(ap + n);
            v2f bb;
            bb.x = bp[(size_t)n * HIDN];
            bb.y = bp[(size_t)n * HIDN + HIDN];
            acc = __builtin_amdgcn_wmma_f32_16x16x4_f32(false, a, false, bb,
                                                        (short)0, acc, false, false);
        }
        if (lane < 16) {  // kh==0 holds D rows 0..7 == heads 0..7
            float* cp = ctx + (size_t)b * NHD * HIDN + j0 + lane;
            #pragma unroll
            for (int r = 0; r < 8; ++r) cp[(size_t)r * HIDN] = acc[r];
        }
    }
}

// ---------------------------------------------------------------------------
// Generic D = A * W^T via fp32 WMMA. One wave -> 16x64 output tile (4 accums).
// ---------------------------------------------------------------------------
__global__ void __launch_bounds__(256)
gemm_nt_wmma(const float* __restrict__ A, int lda,
             const float* __restrict__ W, int ldw,
             float* __restrict__ D, int ldd,
             int M, int Ncols, int K) {
    const int lane = threadIdx.x & 31;
    const int wave = (blockIdx.x * blockDim.x + threadIdx.x) >> 5;
    const int mTiles = M >> 4;
    const int nGroups = Ncols >> 6;
    if (wave >= mTiles * nGroups) return;
    const int mt = wave % mTiles;
    const int ng = wave / mTiles;
    const int m0 = mt << 4, n0 = ng << 6;
    const int row = lane & 15, kh = lane >> 4;

    const float* ap  = A + (size_t)(m0 + row) * lda + 2 * kh;
    const float* wp0 = W + (size_t)(n0 + row) * ldw + 2 * kh;
    const float* wp1 = wp0 + (size_t)16 * ldw;
    const float* wp2 = wp0 + (size_t)32 * ldw;
    const float* wp3 = wp0 + (size_t)48 * ldw;

    v8f acc0 = {}, acc1 = {}, acc2 = {}, acc3 = {};
    for (int k = 0; k < K; k += 4) {
        v2f a  = *(const v2f*)(ap + k);
        v2f b0 = *(const v2f*)(wp0 + k);
        v2f b1 = *(const v2f*)(wp1 + k);
        v2f b2 = *(const v2f*)(wp2 + k);
        v2f b3 = *(const v2f*)(wp3 + k);
        acc0 = __builtin_amdgcn_wmma_f32_16x16x4_f32(false, a, false, b0, (short)0, acc0, false, false);
        acc1 = __builtin_amdgcn_wmma_f32_16x16x4_f32(false, a, false, b1, (short)0, acc1, false, false);
        acc2 = __builtin_amdgcn_wmma_f32_16x16x4_f32(false, a, false, b2, (short)0, acc2, false, false);
        acc3 = __builtin_amdgcn_wmma_f32_16x16x4_f32(false, a, false, b3, (short)0, acc3, false, false);
    }
    float* dp = D + (size_t)(m0 + 8 * kh) * ldd + n0 + row;
    #pragma unroll
    for (int r = 0; r < 8; ++r) {
        dp[(size_t)r * ldd +  0] = acc0[r];
        dp[(size_t)r * ldd + 16] = acc1[r];
        dp[(size_t)r * ldd + 32] = acc2[r];
        dp[(size_t)r * ldd + 48] = acc3[r];
    }
}

// ---------------------------------------------------------------------------
extern "C" void kernel_launch(void* const* d_in, const int* in_sizes, int n_in,
                              void* d_out, int out_size, void* d_ws, size_t ws_size,
                              hipStream_t stream) {
    (void)in_sizes; (void)n_in; (void)out_size; (void)ws_size;
    const float* h        = (const float*)d_in[0];
    const int*   lengths  = (const int*)  d_in[1];
    const float* w_q      = (const float*)d_in[2];
    // d_in[3] = w_k : dead in the reference computation
    const float* w_v      = (const float*)d_in[4];
    const float* w_o      = (const float*)d_in[5];
    const float* w_seed_w = (const float*)d_in[6];
    const float* w_seed_b = (const float*)d_in[7];
    const float* seed_vec = (const float*)d_in[8];

    float* out   = (float*)d_out;                 // [128, 512]
    float* alpha = out + (size_t)NB * HIDN;       // [128, 8, 512, 1]

    float* ws   = (float*)d_ws;
    float* qt   = ws;                             // 8*512            = 4096 floats
    float* ctx  = qt + NHD * HIDN;                // 128*8*512        = 524288 floats
    float* outh = ctx + (size_t)NB * NHD * HIDN;  // 128*512          = 65536 floats

    // 1) tiny prep: q and folded qt
    prep_qt_kernel<<<1, 256, 0, stream>>>(w_q, w_v, w_seed_w, w_seed_b, seed_vec, qt);

    // 2) fused logits + masked softmax + context, one block per batch element
    pma_core_kernel<<<NB, 256, 0, stream>>>(h, lengths, qt, alpha, ctx);

    // 3) per-head: out_h[:, h*64:(h+1)*64] = ctx[:,h,:] @ W_v[h-rows]^T
    for (int hd = 0; hd < NHD; ++hd) {
        gemm_nt_wmma<<<1, 256, 0, stream>>>(
            ctx + (size_t)hd * HIDN, NHD * HIDN,
            w_v + (size_t)hd * DHD * HIDN, HIDN,
            outh + (size_t)hd * DHD, HIDN,
            NB, DHD, HIDN);
    }

    // 4) final projection: out = out_h @ w_o^T   (M=128, N=512, K=512)
    gemm_nt_wmma<<<8, 256, 0, stream>>>(outh, HIDN, w_o, HIDN, out, HIDN,
                                        NB, HIDN, HIDN);
}